// GraphSAGE_37752762532360
// MI455X (gfx1250) — compile-verified
//
#include <hip/hip_runtime.h>
#include <hip/hip_bf16.h>
#include <math.h>

// ---------------------------------------------------------------------------
// GraphSAGE on MI455X (gfx1250, wave32).
//   N=40000 nodes, E=640000 edges, D=H=128, G=64 graphs, C=10 classes, L=3.
// GEMMs (7.9 GFLOP total) use fp32 WMMA V_WMMA_F32_16X16X4_F32 (full
// precision; GEMM is nowhere near the bottleneck). Edge scatter (~2GB of
// L2-resident float atomics) dominates; node features (20.5MB) fit in 192MB L2.
// Weights are repacked into an interleaved [k/2][n][4] layout so each WMMA
// B-fragment pair (Wl k,k+1 / Wr k,k+1) is a single coalesced b128 load.
// ---------------------------------------------------------------------------

#define NN 40000
#define NE 640000
#define DD 128
#define GG 64
#define CC 10
#define LL 3

typedef float v2f __attribute__((ext_vector_type(2)));
typedef float v8f __attribute__((ext_vector_type(8)));

// ---------------- utility kernels ----------------

__global__ void k_zero(float* __restrict__ p, int n) {
    int i = blockIdx.x * blockDim.x + threadIdx.x;
    if (i < n) p[i] = 0.0f;
}

__global__ void k_deg(const int* __restrict__ dst, float* __restrict__ deg, int nE) {
    int e = blockIdx.x * blockDim.x + threadIdx.x;
    if (e < nE) atomicAdd(&deg[dst[e]], 1.0f);
}

__global__ void k_rdeg(float* __restrict__ deg, int n) {
    int i = blockIdx.x * blockDim.x + threadIdx.x;
    if (i < n) deg[i] = 1.0f / fmaxf(deg[i], 1.0f);
}

// Interleaved weight pack: Wpk[l][kp][n] = { Wl[l][n][2kp], Wl[l][n][2kp+1],
//                                            Wr[l][n][2kp], Wr[l][n][2kp+1] }
// -> one b128 load yields both k-pair fragments for both weight matrices.
__global__ void k_pack(const float* __restrict__ Wl, const float* __restrict__ Wr,
                       float4* __restrict__ Wpk, int total /* LL*64*128 */) {
    int i = blockIdx.x * blockDim.x + threadIdx.x;
    if (i >= total) return;
    int l   = i >> 13;           // / (64*128)
    int rem = i & 8191;
    int kp  = rem >> 7;          // 0..63
    int n   = rem & 127;         // 0..127
    const float* wl = Wl + (size_t)l * 16384 + n * 128 + 2 * kp;
    const float* wr = Wr + (size_t)l * 16384 + n * 128 + 2 * kp;
    float4 v;
    v.x = wl[0]; v.y = wl[1];
    v.z = wr[0]; v.w = wr[1];
    Wpk[(size_t)l * 8192 + (size_t)kp * 128 + n] = v;
}

// ---------------- edge scatter: agg[dst] += h[src]  (one wave per edge) ----

__global__ __launch_bounds__(256)
void k_scatter(const int* __restrict__ src, const int* __restrict__ dst,
               const float* __restrict__ h, float* __restrict__ agg, int nE) {
    int gid  = blockIdx.x * blockDim.x + threadIdx.x;
    int e    = gid >> 5;
    int lane = gid & 31;
    if (e >= nE) return;
    int s = src[e];
    int d = dst[e];
    const float4 v = *(const float4*)(h + (size_t)s * DD + lane * 4);
    float* o = agg + (size_t)d * DD + lane * 4;
    atomicAdd(o + 0, v.x);
    atomicAdd(o + 1, v.y);
    atomicAdd(o + 2, v.z);
    atomicAdd(o + 3, v.w);
}

// ---------------- fused SAGE layer GEMM --------------------------------------
// hout = relu( (agg * rdeg) @ Wl^T + bl + hin @ Wr^T )
// One wave per 16-node M-tile, full N=128 in 8 v8f accumulators.
// fp32 WMMA layouts (wave32):
//   A 16x4 : lanes0-15 -> M=lane,  v0=K0,v1=K1 ; lanes16-31 -> v0=K2,v1=K3
//   B 4x16 : v0: lanes0-15 K=0 (N=lane), lanes16-31 K=2 ; v1: K=1 / K=3
//   C 16x16: vgpr r: lanes0-15 (M=r,N=lane), lanes16-31 (M=r+8,N=lane-16)

__global__ __launch_bounds__(160)
void k_gemm_sage(const float* __restrict__ hin, const float* __restrict__ agg,
                 const float* __restrict__ rdeg,
                 const float4* __restrict__ Wpk, const float* __restrict__ bl,
                 float* __restrict__ hout, int nNodes) {
    const int lane  = threadIdx.x & 31;
    const int wave  = threadIdx.x >> 5;
    const int mtile = blockIdx.x * 5 + wave;           // 2500 tiles / 5 waves exact
    const int m0    = mtile * 16;
    if (m0 >= nNodes) return;                          // wave-uniform; never taken here

    const int half = lane >> 4;                        // 0: lanes0-15, 1: lanes16-31
    const int mr   = lane & 15;
    const int row  = m0 + mr;
    const float rd = rdeg[row];

    v8f acc[8];
    // bias: per C/D layout the column index is mr for both lane halves
    for (int nt = 0; nt < 8; ++nt) {
        const float b = bl[nt * 16 + mr];
        for (int r = 0; r < 8; ++r) acc[nt][r] = b;
    }

    const float* aggRow = agg + (size_t)row * DD;
    const float* hRow   = hin + (size_t)row * DD;

    for (int k = 0; k < DD; k += 4) {
        const int ka = k + 2 * half;                   // this half's K pair
        const int kp = ka >> 1;                        // packed k-pair index
        const float2 ta = *(const float2*)(aggRow + ka);
        const float2 th = *(const float2*)(hRow + ka);
        v2f a_agg, a_h;
        a_agg[0] = ta.x * rd;  a_agg[1] = ta.y * rd;
        a_h[0]   = th.x;       a_h[1]   = th.y;

        const float4* wrow = Wpk + (size_t)kp * 128;
        for (int nt = 0; nt < 8; ++nt) {
            const int col = nt * 16 + mr;
            const float4 wv = wrow[col];               // single b128: Wl k,k+1 | Wr k,k+1
            v2f bv, cv;
            bv[0] = wv.x; bv[1] = wv.y;
            cv[0] = wv.z; cv[1] = wv.w;
            acc[nt] = __builtin_amdgcn_wmma_f32_16x16x4_f32(
                false, a_agg, false, bv, (short)0, acc[nt], false, false);
            acc[nt] = __builtin_amdgcn_wmma_f32_16x16x4_f32(
                false, a_h,   false, cv, (short)0, acc[nt], false, false);
        }
    }

    // ReLU epilogue + store (row = r + 8*half, col = nt*16 + mr)
    for (int nt = 0; nt < 8; ++nt) {
        for (int r = 0; r < 8; ++r) {
            const int mm = m0 + r + 8 * half;
            const int cc = nt * 16 + mr;
            hout[(size_t)mm * DD + cc] = fmaxf(acc[nt][r], 0.0f);
        }
    }
}

// ---------------- global mean pool (wave per node) ---------------------------

__global__ __launch_bounds__(256)
void k_pool(const float* __restrict__ h, const int* __restrict__ batch,
            float* __restrict__ g, float* __restrict__ cnt, int n) {
    int gid  = blockIdx.x * blockDim.x + threadIdx.x;
    int node = gid >> 5;
    int lane = gid & 31;
    if (node >= n) return;
    int b = batch[node];
    const float4 v = *(const float4*)(h + (size_t)node * DD + lane * 4);
    float* o = g + (size_t)b * DD + lane * 4;
    atomicAdd(o + 0, v.x);
    atomicAdd(o + 1, v.y);
    atomicAdd(o + 2, v.z);
    atomicAdd(o + 3, v.w);
    if (lane == 0) atomicAdd(&cnt[b], 1.0f);
}

// ---------------- MLP head + log_softmax (one block per graph) ---------------

__global__ __launch_bounds__(128)
void k_head(const float* __restrict__ g, const float* __restrict__ cnt,
            const float* __restrict__ W1, const float* __restrict__ b1,
            const float* __restrict__ W2, const float* __restrict__ b2,
            float* __restrict__ out) {
    __shared__ float gm[128];
    __shared__ float h1[128];
    __shared__ float lg[CC];
    const int gi = blockIdx.x;
    const int t  = threadIdx.x;

    const float c = fmaxf(cnt[gi], 1.0f);
    gm[t] = g[gi * 128 + t] / c;
    __syncthreads();

    float s = b1[t];
    for (int k = 0; k < 128; ++k) s += gm[k] * W1[t * 128 + k];
    h1[t] = fmaxf(s, 0.0f);
    __syncthreads();

    if (t < CC) {
        float s2 = b2[t];
        for (int k = 0; k < 128; ++k) s2 += h1[k] * W2[t * 128 + k];
        lg[t] = s2;
    }
    __syncthreads();

    if (t == 0) {
        float m = lg[0];
        for (int i = 1; i < CC; ++i) m = fmaxf(m, lg[i]);
        float sum = 0.0f;
        for (int i = 0; i < CC; ++i) sum += expf(lg[i] - m);
        const float lse = m + logf(sum);
        for (int i = 0; i < CC; ++i) out[gi * CC + i] = lg[i] - lse;
    }
}

// ---------------- launch -----------------------------------------------------

extern "C" void kernel_launch(void* const* d_in, const int* in_sizes, int n_in,
                              void* d_out, int out_size, void* d_ws, size_t ws_size,
                              hipStream_t stream) {
    const float* x       = (const float*)d_in[0];
    const float* conv_Wl = (const float*)d_in[1];
    const float* conv_bl = (const float*)d_in[2];
    const float* conv_Wr = (const float*)d_in[3];
    const float* lin1_W  = (const float*)d_in[4];
    const float* lin1_b  = (const float*)d_in[5];
    const float* lin2_W  = (const float*)d_in[6];
    const float* lin2_b  = (const float*)d_in[7];
    const int*   ei      = (const int*)d_in[8];    // [2, E] flat
    const int*   batch   = (const int*)d_in[9];
    float*       out     = (float*)d_out;

    const int* srcp = ei;
    const int* dstp = ei + NE;

    // workspace layout (floats): ~62 MB total
    float* w    = (float*)d_ws;
    float* rdeg = w;                       // 40960 (padded)
    float* agg  = rdeg + 40960;            // NN*DD
    float* hb0  = agg  + (size_t)NN * DD;  // NN*DD
    float* hb1  = hb0  + (size_t)NN * DD;  // NN*DD
    float* gbuf = hb1  + (size_t)NN * DD;  // GG*DD
    float* cnt  = gbuf + GG * DD;          // GG
    float4* wpk = (float4*)(cnt + 64);     // LL * 64 * 128 float4 (384 KB)

    const int B = 256;

    // degrees -> reciprocal
    k_zero<<<(NN + B - 1) / B, B, 0, stream>>>(rdeg, NN);
    k_deg<<<(NE + B - 1) / B, B, 0, stream>>>(dstp, rdeg, NE);
    k_rdeg<<<(NN + B - 1) / B, B, 0, stream>>>(rdeg, NN);

    // interleaved weight pack once per call (L2-hot afterwards)
    const int PTOT = LL * 64 * 128;
    k_pack<<<(PTOT + B - 1) / B, B, 0, stream>>>(conv_Wl, conv_Wr, wpk, PTOT);

    const float* hin = x;
    float* houts[LL] = {hb0, hb1, hb0};
    for (int l = 0; l < LL; ++l) {
        k_zero<<<((int)((size_t)NN * DD) + B - 1) / B, B, 0, stream>>>(agg, NN * DD);
        k_scatter<<<(NE * 32 + B - 1) / B, B, 0, stream>>>(srcp, dstp, hin, agg, NE);
        // 2500 M-tiles, 5 waves (160 thr) per block -> 500 blocks, no tail
        k_gemm_sage<<<500, 160, 0, stream>>>(hin, agg, rdeg,
                                             wpk + (size_t)l * 8192,
                                             conv_bl + l * 128, houts[l], NN);
        hin = houts[l];
    }

    k_zero<<<(GG * DD + B - 1) / B, B, 0, stream>>>(gbuf, GG * DD);
    k_zero<<<1, 64, 0, stream>>>(cnt, GG);
    k_pool<<<(NN * 32 + B - 1) / B, B, 0, stream>>>(hin, batch, gbuf, cnt, NN);
    k_head<<<GG, 128, 0, stream>>>(gbuf, cnt, lin1_W, lin1_b, lin2_W, lin2_b, out);
}